// FeatureConsistencyLoss_50251117363410
// MI455X (gfx1250) — compile-verified
//
#include <hip/hip_runtime.h>
#include <math.h>

typedef __attribute__((ext_vector_type(2))) float v2f;
typedef __attribute__((ext_vector_type(8))) float v8f;

#define RADIUS 5
#define DIAM   11
#define HH     128
#define WW     128
#define CF     64
#define CG     3
#define PADW   (WW + 2*RADIUS)   // 138
#define TILEW  144               // 9 tiles of 16, padded
#define LSTR   66                // floats per pixel in LDS (bank-conflict padding)
#define NBLK   (2*HH*DIAM)       // 2816 workgroups in main kernel
#define TOTAL_COUNT (2.0*121.0*128.0*128.0)

__device__ __forceinline__ int refl(int t) {
  if (t < 0) t = -t;
  if (t >= WW) t = 2*WW - 2 - t;
  return t;
}

// Phase 1: per-pixel inverse feature norms -> ws
__global__ void fcl_norms_kernel(const float* __restrict__ feat, float* __restrict__ inorm) {
  int p = blockIdx.x * blockDim.x + threadIdx.x;   // 0..32767
  int b = p >> 14;
  int r = p & 16383;
  const float* fp = feat + (size_t)b * CF * (HH*WW) + r;
  float s = 0.f;
  #pragma unroll 8
  for (int c = 0; c < CF; ++c) { float v = fp[(size_t)c * (HH*WW)]; s += v*v; }
  inorm[p] = 1.0f / fmaxf(sqrtf(s), 1e-8f);
}

// Phase 2: one workgroup per (b, y, dy). 8 waves, each owns a 16-pixel center tile.
__global__ __launch_bounds__(256)
void fcl_main_kernel(const float* __restrict__ guid, const float* __restrict__ feat,
                     const float* __restrict__ inorm, float* __restrict__ partials) {
  __shared__ float ldsC[TILEW*LSTR];   // center row, [px][c], reflect-padded in x
  __shared__ float ldsN[TILEW*LSTR];   // neighbor row (reflected y+dy)
  __shared__ float gC[CG][WW];
  __shared__ float gN[CG][TILEW];
  __shared__ float inC[WW];
  __shared__ float inN[TILEW];
  __shared__ float red[8];

  int bid = blockIdx.x;
  int dyi = bid % DIAM;
  int y   = (bid / DIAM) % HH;
  int b   = bid / (DIAM*HH);
  int ry  = refl(y + dyi - RADIUS);
  int tid = threadIdx.x;

  const float* fb = feat + (size_t)b * CF * HH * WW;
  const float* gb = guid + (size_t)b * CG * HH * WW;

  // Stage features (pixel-major, padded to TILEW with zeros)
  for (int t = tid; t < TILEW*CF; t += 256) {
    int c  = t / TILEW;
    int px = t - c*TILEW;
    float vc = 0.f, vn = 0.f;
    if (px < PADW) {
      int rx = refl(px - RADIUS);
      vc = fb[((size_t)c*HH + y )*WW + rx];
      vn = fb[((size_t)c*HH + ry)*WW + rx];
    }
    ldsC[px*LSTR + c] = vc;
    ldsN[px*LSTR + c] = vn;
  }
  // Stage guidance rows
  for (int t = tid; t < CG*TILEW; t += 256) {
    int c  = t / TILEW;
    int px = t - c*TILEW;
    float vn = 0.f;
    if (px < PADW) vn = gb[((size_t)c*HH + ry)*WW + refl(px - RADIUS)];
    gN[c][px] = vn;
    if (px < WW) gC[c][px] = gb[((size_t)c*HH + y)*WW + px];
  }
  // Stage inverse norms
  for (int t = tid; t < TILEW; t += 256) {
    inN[t] = (t < PADW) ? inorm[((size_t)b*HH + ry)*WW + refl(t - RADIUS)] : 0.f;
    if (t < WW) inC[t] = inorm[((size_t)b*HH + y)*WW + t];
  }
  __syncthreads();

  int wave = tid >> 5;
  int lane = tid & 31;
  int m    = lane & 15;            // M (A) / N (B) index for this lane
  int kq   = (lane >> 4) << 1;     // K sub-offset {0,2} per half-wave (ISA f32 16x16x4 layout)
  int x0   = wave << 4;            // center tile base (unpadded x)
  const float* aBase = &ldsC[(x0 + RADIUS + m)*LSTR + kq];
  float dyf = (float)(dyi - RADIUS);

  float local = 0.f;
  #pragma unroll
  for (int t = 0; t < 2; ++t) {
    int j0 = x0 + 16*t;            // neighbor tile base (padded coords)
    const float* bBase = &ldsN[(j0 + m)*LSTR + kq];
    v8f acc = {};
    #pragma unroll
    for (int kb = 0; kb < CF; kb += 4) {
      v2f a  = *(const v2f*)(aBase + kb);
      v2f bv = *(const v2f*)(bBase + kb);
      // D = A(16x4 centers) x B(4x16 neighbors) + C, full f32
      acc = __builtin_amdgcn_wmma_f32_16x16x4_f32(false, a, false, bv,
                                                  (short)0, acc, false, false);
    }
    // Extract the valid diagonal band: dx+5 = 16t + n - i in [0,10]
    #pragma unroll
    for (int r = 0; r < 8; ++r) {
      int i   = r + ((lane >> 4) << 3);   // M = r (lanes 0-15) or r+8 (lanes 16-31)
      int dxp = 16*t + m - i;
      if (dxp >= 0 && dxp <= 10) {
        int x  = x0 + i;                  // center x (unpadded)
        int px = x + dxp;                 // neighbor padded col ( == j0 + m )
        float sim = acc[r] * inC[x] * inN[px];
        float fd  = fmaxf(1.0f - sim, 0.0f);
        float dxf = (float)(dxp - RADIUS);
        float coord = (dyf*dyf + dxf*dxf) * 0.04f;   // /25
        float gd = 0.f;
        #pragma unroll
        for (int c = 0; c < CG; ++c) { float d = gN[c][px] - gC[c][x]; gd += d*d; }
        // W1*exp(-coord - 5*gdiff) + W2*exp(-coord) - SHIFT
        float kern = __expf(-coord - 5.0f*gd) + 0.5f*__expf(-coord) - 0.1f;
        local += fd * kern;
      }
    }
  }

  // wave32 reduce, then block reduce, deterministic partial per workgroup
  #pragma unroll
  for (int off = 16; off > 0; off >>= 1)
    local += __shfl_down(local, off, 32);
  if (lane == 0) red[wave] = local;
  __syncthreads();
  if (tid == 0) {
    float s = 0.f;
    #pragma unroll
    for (int w = 0; w < 8; ++w) s += red[w];
    partials[bid] = s;
  }
}

// Phase 3: deterministic tree reduction of 2816 partials -> mean
__global__ void fcl_reduce_kernel(const float* __restrict__ partials, float* __restrict__ out) {
  __shared__ float sb[256];
  float s = 0.f;
  for (int i = threadIdx.x; i < NBLK; i += 256) s += partials[i];
  sb[threadIdx.x] = s;
  __syncthreads();
  for (int o = 128; o > 0; o >>= 1) {
    if ((int)threadIdx.x < o) sb[threadIdx.x] += sb[threadIdx.x + o];
    __syncthreads();
  }
  if (threadIdx.x == 0) out[0] = sb[0] * (float)(1.0 / TOTAL_COUNT);
}

extern "C" void kernel_launch(void* const* d_in, const int* in_sizes, int n_in,
                              void* d_out, int out_size, void* d_ws, size_t ws_size,
                              hipStream_t stream) {
  const float* guidance = (const float*)d_in[0];  // (2,3,128,128) f32
  const float* features = (const float*)d_in[1];  // (2,64,128,128) f32
  float* inorm    = (float*)d_ws;                 // 2*128*128 floats
  float* partials = inorm + 2*HH*WW;              // NBLK floats

  fcl_norms_kernel<<<(2*HH*WW)/256, 256, 0, stream>>>(features, inorm);
  fcl_main_kernel<<<NBLK, 256, 0, stream>>>(guidance, features, inorm, partials);
  fcl_reduce_kernel<<<1, 256, 0, stream>>>(partials, (float*)d_out);
}